// BezierSurfaceFitter_4294967296014
// MI455X (gfx1250) — compile-verified
//
#include <hip/hip_runtime.h>

typedef __attribute__((ext_vector_type(2))) float v2f;
typedef __attribute__((ext_vector_type(8))) float v8f;

// ---------------------------------------------------------------------------
// Bernstein basis table: bu[i][p] = C(31,p) * t^p * (1-t)^(31-p), t=(i+0.5)/256
// Computed in double (f32 underflows at (1-t)^31 ~ 1e-84), stored as f32.
// H==W and M==N so one table serves both u and v directions.
// ---------------------------------------------------------------------------
__global__ void bernstein_basis_kernel(float* __restrict__ bu) {
    const int i = threadIdx.x;            // 0..255
    double t   = (i + 0.5) / 256.0;
    double omt = 1.0 - t;
    double r   = t / omt;
    double v   = 1.0;
    for (int k = 0; k < 31; ++k) v *= omt;     // (1-t)^31
    for (int p = 0; p < 32; ++p) {
        bu[i * 32 + p] = (float)v;
        v *= r * (double)(31 - p) / (double)(p + 1);
    }
}

// ---------------------------------------------------------------------------
// One workgroup (8 wave32) per (b,c) slice:
//   Stage 1: Tsh(32x256) = Kslice(32x32) * BV^T(32x256)   [BV^T[q][j]=BU[j][q]]
//   Stage 2: out_slice(256x256) = BU(256x32) * Tsh(32x256)
// Both stages use V_WMMA_F32_16X16X4_F32, K-loop of 8 per 16x16 tile.
// All tile loops have compile-time trip counts -> EXEC stays all-ones
// (WMMA requirement) and no per-lane loop masking is generated.
// ---------------------------------------------------------------------------
__launch_bounds__(256)
__global__ void bezier_wmma_kernel(const float* __restrict__ Kmat,
                                   const float* __restrict__ BU,
                                   float* __restrict__ out) {
    __shared__ float BUsh[256 * 32];   // 32 KB basis table
    __shared__ float Ksh[32 * 32];     //  4 KB control points (this slice)
    __shared__ float Tsh[32 * 256];    // 32 KB intermediate

    const int tid  = threadIdx.x;
    const int wave = tid >> 5;
    const int lane = tid & 31;
    const int s    = blockIdx.x;       // slice = b*3 + c

    // cooperative LDS fills (coalesced b32)
    for (int idx = tid; idx < 256 * 32; idx += 256) BUsh[idx] = BU[idx];
    {
        const float* Ks = Kmat + s * 1024;
        for (int idx = tid; idx < 1024; idx += 256) Ksh[idx] = Ks[idx];
    }
    __syncthreads();

    // WMMA f32 16x16x4 per-lane indices (ISA 7.12.2 layouts):
    //  A: lane L, vgpr j  -> A[L&15][2*(L>>4)+j]
    //  B: lane L, vgpr j  -> B[2*(L>>4)+j][L&15]
    //  C/D: lane L, vgpr r -> D[r + 8*(L>>4)][L&15]
    const int m   = lane & 15;
    const int kb  = (lane >> 4) << 1;
    const int hi8 = (lane >> 4) << 3;

    // ---- Stage 1: T = K * BV^T (2x16 = 32 tiles, exactly 4 per wave) ----
#pragma unroll
    for (int it = 0; it < 4; ++it) {
        const int tt  = wave + it * 8;
        const int p0t = (tt >> 4) * 16;    // T row base (0 or 16)
        const int j0  = (tt & 15) * 16;    // T col base
        v8f acc = {0.f, 0.f, 0.f, 0.f, 0.f, 0.f, 0.f, 0.f};
#pragma unroll
        for (int q0 = 0; q0 < 32; q0 += 4) {
            v2f a, b;
            a.x = Ksh[(p0t + m) * 32 + q0 + kb];
            a.y = Ksh[(p0t + m) * 32 + q0 + kb + 1];
            // B[k][n] = BV^T[q0+k][j0+n] = BU[(j0+n)*32 + q0+k]
            b.x = BUsh[(j0 + m) * 32 + q0 + kb];
            b.y = BUsh[(j0 + m) * 32 + q0 + kb + 1];
            acc = __builtin_amdgcn_wmma_f32_16x16x4_f32(
                false, a, false, b, (short)0, acc, false, false);
        }
#pragma unroll
        for (int r = 0; r < 8; ++r)
            Tsh[(p0t + r + hi8) * 256 + j0 + m] = acc[r];
    }
    __syncthreads();

    // ---- Stage 2: out = BU * T (16x16 = 256 tiles, exactly 32 per wave) ----
    float* outs = out + (size_t)s * 65536;
#pragma unroll 1
    for (int it = 0; it < 32; ++it) {
        const int tt = wave + it * 8;
        const int i0 = (tt >> 4) * 16;
        const int j0 = (tt & 15) * 16;
        v8f acc = {0.f, 0.f, 0.f, 0.f, 0.f, 0.f, 0.f, 0.f};
#pragma unroll
        for (int p0 = 0; p0 < 32; p0 += 4) {
            v2f a, b;
            a.x = BUsh[(i0 + m) * 32 + p0 + kb];
            a.y = BUsh[(i0 + m) * 32 + p0 + kb + 1];
            b.x = Tsh[(p0 + kb) * 256 + j0 + m];
            b.y = Tsh[(p0 + kb + 1) * 256 + j0 + m];
            acc = __builtin_amdgcn_wmma_f32_16x16x4_f32(
                false, a, false, b, (short)0, acc, false, false);
        }
#pragma unroll
        for (int r = 0; r < 8; ++r)
            outs[(i0 + r + hi8) * 256 + j0 + m] = acc[r];
    }
}

extern "C" void kernel_launch(void* const* d_in, const int* in_sizes, int n_in,
                              void* d_out, int out_size, void* d_ws, size_t ws_size,
                              hipStream_t stream) {
    // d_in[0] = x (ignored int placeholder), d_in[1] = K_mat f32 (256*3*32*32)
    const float* Kmat = (const float*)d_in[1];
    float*       out  = (float*)d_out;        // (1,256,3,256,256) f32
    float*       BU   = (float*)d_ws;         // 256*32 f32 basis table (32 KB)

    bernstein_basis_kernel<<<1, 256, 0, stream>>>(BU);
    bezier_wmma_kernel<<<768, 256, 0, stream>>>(Kmat, BU, out);
}